// GCN_layer_12678743458315
// MI455X (gfx1250) — compile-verified
//
#include <hip/hip_runtime.h>

typedef __attribute__((ext_vector_type(2))) float v2f;
typedef __attribute__((ext_vector_type(8))) float v8f;

#define N_NODES 10000
#define N_EDGES 160000
#define DIM     512
#define P_DROP  0.3f
#define A_PITCH (DIM + 4)   // 516 floats: bank stride 4 mod 64 -> conflict-free

// ---------------------------------------------------------------------------
// Kernel 1: per-node neighbor aggregation. Edges are sorted by dst, so each
// block binary-searches its node's edge range; (src, 1/norm) pairs are staged
// in LDS and every thread accumulates 2 features (float2) across the edges.
// No atomics, fully coalesced feature reads (rows are L2-resident: 20.5 MB).
// ---------------------------------------------------------------------------
__global__ __launch_bounds__(256) void gcn_aggregate_kernel(
    const float* __restrict__ nodes,
    const int*   __restrict__ edge_src,
    const int*   __restrict__ edge_dst,
    const float* __restrict__ edge_norm,
    const float* __restrict__ degrees,
    float*       __restrict__ agg)
{
    const int node = blockIdx.x;
    const int t    = threadIdx.x;

    // lower_bound(edge_dst, node) and lower_bound(edge_dst, node+1)
    int lo = 0, hi = N_EDGES;
    while (lo < hi) { int mid = (lo + hi) >> 1; if (edge_dst[mid] < node) lo = mid + 1; else hi = mid; }
    const int e0 = lo;
    hi = N_EDGES;
    while (lo < hi) { int mid = (lo + hi) >> 1; if (edge_dst[mid] < node + 1) lo = mid + 1; else hi = mid; }
    const int e1 = lo;

    __shared__ int   s_src[256];
    __shared__ float s_rnorm[256];

    const float invdeg = 1.0f / degrees[node];
    const float2* nrow = (const float2*)(nodes + (size_t)node * DIM);
    float2 self = nrow[t];
    float accx = self.x * invdeg;
    float accy = self.y * invdeg;

    for (int base = e0; base < e1; base += 256) {
        const int cnt = (e1 - base < 256) ? (e1 - base) : 256;
        __syncthreads();
        if (t < cnt) {
            s_src[t]   = edge_src[base + t];
            s_rnorm[t] = 1.0f / edge_norm[base + t];
        }
        __syncthreads();
        for (int i = 0; i < cnt; ++i) {
            const float2* srow = (const float2*)(nodes + (size_t)s_src[i] * DIM);
            float2 v = srow[t];
            float  r = s_rnorm[i];
            accx = fmaf(v.x, r, accx);
            accy = fmaf(v.y, r, accy);
        }
    }

    float2* arow = (float2*)(agg + (size_t)node * DIM);
    float2 res; res.x = accx; res.y = accy;
    arow[t] = res;
}

// ---------------------------------------------------------------------------
// Kernel 2: out = dropout(relu(agg @ W^T)) via fp32 WMMA (16x16x4).
// Each wave (wave32) owns one 16x16 output tile; 8 waves/block share one
// LDS-staged A tile (16 rows x 512 K, pitch 516 floats). B[k][n] = W[n][k]
// is read directly from L2 (W = 1 MB, hot). N=625*16 rows, D=32*16 cols.
//
// ISA VGPR layouts (CDNA5 ISA):
//   A 16x4 f32 : lane -> m = lane&15, v0/v1 hold K = (lane>>4)*2 + {0,1}
//   B  4x16 f32: lane -> n = lane&15, v0/v1 hold K = (lane>>4)*2 + {0,1}
//   C/D 16x16  : vgpr r, lane l -> M = r + (l>>4)*8, N = l&15
// ---------------------------------------------------------------------------
__global__ __launch_bounds__(256) void gcn_gemm_wmma_kernel(
    const float* __restrict__ agg,
    const float* __restrict__ W,       // [D][D] row-major, W[j][k]
    const float* __restrict__ drop_u,  // [N][D]
    float*       __restrict__ out)     // [N][D]
{
    __shared__ float sA[16 * A_PITCH];

    const int row0  = blockIdx.x * 16;                   // 0..624 tiles of M
    const int wave  = threadIdx.x >> 5;                  // 8 waves per block
    const int lane  = threadIdx.x & 31;
    const int col0  = (blockIdx.y * 8 + wave) * 16;      // 32 tiles of N

    // Cooperative A-tile load: 16 rows x 512 floats = 2048 float4s, 256 thr.
    {
        const int tid = threadIdx.x;
        #pragma unroll
        for (int i = 0; i < 8; ++i) {
            int idx = tid + i * 256;        // float4 index in tile
            int r   = idx >> 7;             // 128 float4 per row
            int c   = (idx & 127) << 2;     // float column
            float4 v = *(const float4*)(agg + (size_t)(row0 + r) * DIM + c);
            float* dst = &sA[r * A_PITCH + c];
            dst[0] = v.x; dst[1] = v.y; dst[2] = v.z; dst[3] = v.w;
        }
    }
    __syncthreads();

    const int mr = lane & 15;             // A row / B col for this lane
    const int kb = (lane >> 4) * 2;       // lane-group K offset within K=4 step

    v8f c = {0.f, 0.f, 0.f, 0.f, 0.f, 0.f, 0.f, 0.f};

    const float* wrow = W + (size_t)(col0 + mr) * DIM;   // B[k][n] = W[col0+n][k]
    const float* arow = &sA[mr * A_PITCH];

    #pragma unroll 4
    for (int k0 = 0; k0 < DIM; k0 += 4) {
        v2f a, b;
        a[0] = arow[k0 + kb];
        a[1] = arow[k0 + kb + 1];
        b[0] = wrow[k0 + kb];
        b[1] = wrow[k0 + kb + 1];
        // 8 args: (neg_a, A, neg_b, B, c_mod, C, reuse_a, reuse_b)
        c = __builtin_amdgcn_wmma_f32_16x16x4_f32(
                false, a, false, b, (short)0, c, false, false);
    }

    // Epilogue: ReLU + inverted dropout, D-matrix layout scatter.
    const float scale = 1.0f / (1.0f - P_DROP);
    const int nn = col0 + (lane & 15);
    #pragma unroll
    for (int r = 0; r < 8; ++r) {
        const int mm = row0 + r + (lane >> 4) * 8;
        float h = c[r];
        h = fmaxf(h, 0.0f);
        const float u    = drop_u[(size_t)mm * DIM + nn];
        const float keep = (u >= P_DROP) ? scale : 0.0f;
        out[(size_t)mm * DIM + nn] = h * keep;
    }
}

// ---------------------------------------------------------------------------
extern "C" void kernel_launch(void* const* d_in, const int* in_sizes, int n_in,
                              void* d_out, int out_size, void* d_ws, size_t ws_size,
                              hipStream_t stream)
{
    const float* nodes     = (const float*)d_in[0];
    const int*   edge_src  = (const int*)  d_in[1];
    const int*   edge_dst  = (const int*)  d_in[2];
    const float* edge_norm = (const float*)d_in[3];
    const float* degrees   = (const float*)d_in[4];
    const float* W         = (const float*)d_in[5];
    const float* drop_u    = (const float*)d_in[6];
    float*       out       = (float*)d_out;
    float*       agg       = (float*)d_ws;   // N*D floats = 20.48 MB scratch

    gcn_aggregate_kernel<<<dim3(N_NODES), dim3(256), 0, stream>>>(
        nodes, edge_src, edge_dst, edge_norm, degrees, agg);

    gcn_gemm_wmma_kernel<<<dim3(N_NODES / 16, 32 / 8), dim3(256), 0, stream>>>(
        agg, W, drop_u, out);
}